// Pooling_87239375716838
// MI455X (gfx1250) — compile-verified
//
#include <hip/hip_runtime.h>

// ---------------------------------------------------------------------------
// Sliding-window 4-point FFT pooling (real part), MI455X / gfx1250.
//
// coarse[i,d] = x[max(0,i-1), d]           (mean of Re(FFT4) collapses to w0)
// fine[i, s*D+d] = [w0+w1+w2+w3, w0-w2, w0-w1+w2-w3, w0-w2][s]
//
// Memory-bound: 64 MB read + 320 MB written ~= 384 MB @ 23.3 TB/s -> ~16.5 us.
// Strategy: async-stage input rows into LDS once (HBM read 1x, ASYNCcnt path),
// compute from LDS, stream the 320 MB of output with non-temporal b128 stores.
// ---------------------------------------------------------------------------

typedef float v4f __attribute__((ext_vector_type(4)));
typedef int   v4i __attribute__((vector_size(16)));   // matches builtin pointee

#ifndef __has_builtin
#define __has_builtin(x) 0
#endif

#if __has_builtin(__builtin_amdgcn_global_load_async_to_lds_b128)
#define USE_ASYNC_LDS 1
typedef __attribute__((address_space(1))) v4i* gptr_t;  // global int4*
typedef __attribute__((address_space(3))) v4i* lptr_t;  // LDS int4*
#else
#define USE_ASYNC_LDS 0
#endif

constexpr int D_DIM    = 128;        // feature dim
constexpr int R_ROWS   = 16;         // output rows per block
constexpr int L_ROWS   = R_ROWS + 3; // staged rows: b0-1 .. b0+R+1 (clamped)
constexpr int C4       = D_DIM / 4;  // 32 float4 chunks per row
constexpr int NTHREADS = 256;

__global__ __launch_bounds__(NTHREADS) void fftpool_kernel(
    const float* __restrict__ x, float* __restrict__ out, int T) {
  __shared__ float lds[L_ROWS * D_DIM];

  const int t  = threadIdx.x;
  const int b0 = blockIdx.x * R_ROWS;

  // ---- Stage rows clamp(b0-1+s, [0,T-1]) for s in [0, L_ROWS) into LDS ----
  // L_ROWS * C4 = 608 b128 transfers, 256 threads -> 3 strided passes.
#pragma unroll
  for (int base = 0; base < L_ROWS * C4; base += NTHREADS) {
    const int chunk = base + t;
    if (chunk < L_ROWS * C4) {
      const int s = chunk >> 5;          // LDS row
      const int c = (chunk & 31) << 2;   // float offset within row
      int g = b0 - 1 + s;                // global row, clamped (masking is
      g = g < 0 ? 0 : (g >= T ? T - 1 : g);  // applied at compute time)
      const float* src = x + (size_t)g * D_DIM + c;
      float*       dst = &lds[s * D_DIM + c];
#if USE_ASYNC_LDS
      __builtin_amdgcn_global_load_async_to_lds_b128(
          (gptr_t)src, (lptr_t)dst, /*ioffset=*/0, /*cpol=*/0);
#else
      *(v4f*)dst = *(const v4f*)src;     // global_load_b128 + ds_store_b128
#endif
    }
  }
#if USE_ASYNC_LDS
#if __has_builtin(__builtin_amdgcn_s_wait_asynccnt)
  __builtin_amdgcn_s_wait_asynccnt(0);
#else
  asm volatile("s_wait_asynccnt 0" ::: "memory");
#endif
#endif
  __syncthreads();

  const size_t TD = (size_t)T * D_DIM;
  const v4f vzero = {0.f, 0.f, 0.f, 0.f};

  // ---- Compute + streaming stores: R_ROWS*C4 = 512 items, 2 per thread ----
#pragma unroll
  for (int base = 0; base < R_ROWS * C4; base += NTHREADS) {
    const int w  = base + t;
    const int li = w >> 5;               // local row 0..15
    const int c  = (w & 31) << 2;        // float offset within row
    const int i  = b0 + li;              // output row
    if (i >= T) continue;

    const bool first = (i == 0);
    // LDS row holding w0 = x[max(0,i-1)]:
    //   i>0: staged row li holds global row i-1; i==0: staged row 1 holds row 0.
    const int sb = first ? 1 : li;
    const float* p = &lds[sb * D_DIM + c];

    v4f w0 = *(const v4f*)(p);
    v4f w1 = *(const v4f*)(p + D_DIM);
    v4f w2 = *(const v4f*)(p + 2 * D_DIM);
    v4f w3 = *(const v4f*)(p + 3 * D_DIM);

    // Right-pad masking (reference: pos < i+3 && pos < T).
    if (first) {
      w3 = vzero;                        // window = [x0,x1,x2,0]
    } else {
      if (i + 1 >= T) w2 = vzero;
      if (i + 2 >= T) w3 = vzero;
    }

    const v4f f0 = (w0 + w1) + (w2 + w3);   // Re F[0]
    const v4f f1 = w0 - w2;                 // Re F[1] == Re F[3]
    const v4f f2 = (w0 - w1) + (w2 - w3);   // Re F[2]
    // coarse = mean(Re F) = w0 exactly.

    float* co = out + (size_t)i * D_DIM + c;
    float* fi = out + TD + (size_t)i * (4 * D_DIM) + c;
    __builtin_nontemporal_store(w0, (v4f*)co);
    __builtin_nontemporal_store(f0, (v4f*)(fi));
    __builtin_nontemporal_store(f1, (v4f*)(fi + D_DIM));
    __builtin_nontemporal_store(f2, (v4f*)(fi + 2 * D_DIM));
    __builtin_nontemporal_store(f1, (v4f*)(fi + 3 * D_DIM));
  }
}

extern "C" void kernel_launch(void* const* d_in, const int* in_sizes, int n_in,
                              void* d_out, int out_size, void* d_ws, size_t ws_size,
                              hipStream_t stream) {
  const float* x = (const float*)d_in[0];
  float* out = (float*)d_out;
  const int T = in_sizes[0] / D_DIM;               // B=1, D=128
  const int grid = (T + R_ROWS - 1) / R_ROWS;      // 8192 blocks for T=131072
  fftpool_kernel<<<grid, NTHREADS, 0, stream>>>(x, out, T);
}